// TemporalGNN_74869869904433
// MI455X (gfx1250) — compile-verified
//
#include <hip/hip_runtime.h>
#include <hip/hip_bf16.h>
#include <math.h>

// Problem constants from the reference
#define TB   32   // batch
#define TFIN 2    // input features
#define TT   12   // T_in (== T_out)
#define TC   8    // channels
#define TOUT 12
#define FT   24   // TFIN*TT floats per (node,batch)
#define NBF  768  // TB*FT floats per node

typedef float v2f __attribute__((ext_vector_type(2)));
typedef float v8f __attribute__((ext_vector_type(8)));

// ---------------- degree / normalization ----------------
__global__ void k_init_deg(float* deg, int n) {
  int i = blockIdx.x * blockDim.x + threadIdx.x;
  if (i < n) deg[i] = 1.0f;  // self-loop fill value
}

__global__ void k_edge_deg(const long long* ei, const float* w, float* deg, int E) {
  int e = blockIdx.x * blockDim.x + threadIdx.x;
  if (e < E) {
    int c = (int)ei[(long)E + e];          // col
    float ww = w[e];
    ww = ww > 0.f ? ww : 0.f;              // relu(edge_weight)
    atomicAdd(&deg[c], ww);
  }
}

__global__ void k_dinv(float* deg, int n) {
  int i = blockIdx.x * blockDim.x + threadIdx.x;
  if (i < n) deg[i] = rsqrtf(deg[i]);      // in-place: deg -> dinv
}

// ---------------- fold weights, softmax(attention) ----------------
// cst layout (floats): [0..15] probs, [16..79] Bmat1 (4x16), [80..207] Bmat2 (8x16)
__global__ void k_precompute(const float* att,
                             const float* Wz, const float* bz,
                             const float* Wh, const float* bh,
                             const float* Lz, const float* Lzb,
                             const float* Lh, const float* Lhb,
                             const float* ow, float* cst) {
  int tid = threadIdx.x;  // 64 threads
  if (tid == 0) {
    float m = att[0];
    for (int i = 1; i < TT; i++) m = fmaxf(m, att[i]);
    float e[TT]; float s = 0.f;
    for (int i = 0; i < TT; i++) { e[i] = __expf(att[i] - m); s += e[i]; }
    for (int i = 0; i < TT; i++) cst[i] = e[i] / s;
    for (int i = TT; i < 16; i++) cst[i] = 0.f;
  }
  // Bmat1[k][ncol]: ncol<8 -> Z path, ncol>=8 -> H path.
  // k=0,1: (W @ L_top)[k][c];  k=2: (b @ L_top + Lb)[c];  k=3: 0
  {
    int k = tid >> 4, ncl = tid & 15;
    const float* W  = (ncl < 8) ? Wz  : Wh;
    const float* L  = (ncl < 8) ? Lz  : Lh;
    const float* Lb = (ncl < 8) ? Lzb : Lhb;
    const float* bb = (ncl < 8) ? bz  : bh;
    int c = ncl & 7;
    float v;
    if (k < 2) {
      v = 0.f;
      for (int j = 0; j < TC; j++) v += W[k * TC + j] * L[j * TC + c];  // first C rows of L
    } else if (k == 2) {
      v = Lb[c];
      for (int j = 0; j < TC; j++) v += bb[j] * L[j * TC + c];
    } else {
      v = 0.f;
    }
    cst[16 + k * 16 + ncl] = v;
  }
  // Bmat2[k][ncol] = out_w[k][ncol] (8x12 padded to 8x16)
  for (int i = tid; i < 128; i += 64) {
    int k = i >> 4, ncl = i & 15;
    cst[80 + i] = (ncl < TOUT) ? ow[k * TOUT + ncl] : 0.f;
  }
}

// ---------------- Xagg = snorm*x (self term) ----------------
// Xagg layout: [(node*TB + b)*FT + ft], ft = f*TT + t
__global__ void k_init_xagg(const float* x, const float* dinv, float* xagg, int n) {
  long i = (long)blockIdx.x * blockDim.x + threadIdx.x;
  long tot = (long)n * NBF;
  if (i < tot) {
    int node = (int)(i / NBF);
    int q    = (int)(i - (long)node * NBF);
    int b    = q / FT;
    int ft   = q - b * FT;
    float sn = dinv[node]; sn = sn * sn;                 // snorm = dinv^2
    xagg[i] = sn * x[(long)b * n * FT + (long)node * FT + ft];
  }
}

// ---------------- edge scatter: one wave per edge ----------------
// Per edge: add enorm * x[row, :, :, :] into Xagg[col]. 768 floats, linear
// index q = b*FT + ft is contiguous in BOTH source (within a b-segment) and
// destination -> coalesced loads and coalesced float atomics (L2-resident).
__global__ void k_edge_scatter(const long long* ei, const float* w, const float* dinv,
                               const float* x, float* xagg, int E, int n) {
  int wid  = (int)(((long)blockIdx.x * blockDim.x + threadIdx.x) >> 5);
  int lane = threadIdx.x & 31;
  if (wid >= E) return;
  int r = (int)ei[wid];
  int c = (int)ei[(long)E + wid];
  float ww = w[wid]; ww = ww > 0.f ? ww : 0.f;
  float en = dinv[r] * ww * dinv[c];
  if (en == 0.f) return;                                  // wave-uniform
  float* dst = xagg + (long)c * NBF;
  long srow = (long)r * FT;
  long strideB = (long)n * FT;
#pragma unroll 4
  for (int i = 0; i < 24; i++) {                          // 24*32 = 768
    int q  = i * 32 + lane;
    int b  = q / FT;
    int ft = q - b * FT;
    float v = x[(long)b * strideB + srow + ft];
    atomicAdd(&dst[q], en * v);
  }
}

// Branchless gate: Z-lanes need sigmoid(-z) = 0.5 - 0.5*tanh(0.5*z),
// H-lanes need tanh(h). One tanh evaluation covers both.
__device__ __forceinline__ float gate_fn(float val, bool isZ) {
#if __has_builtin(__builtin_amdgcn_tanhf)
  float u  = val * (isZ ? 0.5f : 1.0f);
  float t  = __builtin_amdgcn_tanhf(u);          // native v_tanh_f32
  float aa = isZ ? -0.5f : 1.0f;
  float bb = isZ ?  0.5f : 0.0f;
  return fmaf(aa, t, bb);
#else
  // tanh(u) = 1 - 2/(e^{2u}+1); with u as above, branchless exp-based path
  float u  = val * (isZ ? 1.0f : 2.0f);          // Z: e^{z}; H: e^{2h}
  float e  = __expf(u);
  float g  = __fdividef(1.0f, 1.0f + e);         // Z: sigmoid(-z); H: sigmoid(-2h)
  float aa = isZ ? 1.0f : -2.0f;
  float bb = isZ ? 0.0f :  1.0f;
  return fmaf(aa, g, bb);                        // Z: g ; H: 1-2g = tanh(h)
#endif
}

// ---------------- WMMA cell + output projection ----------------
// One wave handles 16 samples (sample = node*32 + b).
// Stage 1 (12 static WMMAs): D = A(16x4) * B1(4x16), A rows = [f0, f1, 1, 0],
//   B1 cols = [Z_pre(8) | H_pre(8)] with bias folded into the k=2 row. Each
//   lane applies its own gate nonlinearity, then one shfl_xor(8) pairs
//   Z-lane x H-lane:  acc += probs[t] * gate(Z) * gate(H).
// Stage 2: out = relu(acc)(16x8) @ out_w(8x16pad) as two K=4 WMMAs via LDS
//   transpose, + out_b.
__global__ void __launch_bounds__(256)
k_cell(const float* __restrict__ xagg, const float* __restrict__ cst,
       const float* __restrict__ ob, float* __restrict__ out, int n) {
  __shared__ float Hsh[8][16][9];   // [wave][sample row][channel], padded

  int lane = threadIdx.x & 31;
  int wv   = threadIdx.x >> 5;
  long tile = (long)blockIdx.x * 8 + wv;
  long base = tile * 16;
  long total = (long)n * TB;
  bool valid = (base + 16 <= total);

  int  ncol = lane & 15;
  bool hi   = lane >= 16;
  bool isZ  = (lane & 8) == 0;

  if (valid) {
    // B matrix for stage 1 (f32 WMMA B layout: vgpr0 = K{0,2}, vgpr1 = K{1,3})
    const float* B1 = cst + 16;
    v2f b1;
    b1.x = B1[(hi ? 2 : 0) * 16 + ncol];
    b1.y = B1[(hi ? 3 : 1) * 16 + ncol];

    // hoist attention probs (uniform -> scalar regs)
    float pr[TT];
#pragma unroll
    for (int t = 0; t < TT; t++) pr[t] = cst[t];

    // preload this lane's 24 features as 6 x b128 (96 B, 16-B aligned)
    float arr[FT];
    if (!hi) {
      const float4* xp = (const float4*)(xagg + (base + lane) * FT);
#pragma unroll
      for (int i = 0; i < 6; i++) {
        float4 v = xp[i];
        arr[4 * i + 0] = v.x; arr[4 * i + 1] = v.y;
        arr[4 * i + 2] = v.z; arr[4 * i + 3] = v.w;
      }
    } else {
#pragma unroll
      for (int i = 0; i < FT; i++) arr[i] = 0.f;
    }

    float acc[8] = {0.f, 0.f, 0.f, 0.f, 0.f, 0.f, 0.f, 0.f};

#pragma unroll
    for (int t = 0; t < TT; t++) {
      v2f a;
      a.x = hi ? 1.0f : arr[t];        // lanes 0-15: K0=f0(t); 16-31: K2=1 (bias)
      a.y = hi ? 0.0f : arr[t + TT];   // lanes 0-15: K1=f1(t); 16-31: K3=0
      v8f cz = {0.f, 0.f, 0.f, 0.f, 0.f, 0.f, 0.f, 0.f};
      v8f d = __builtin_amdgcn_wmma_f32_16x16x4_f32(
          false, a, false, b1, (short)0, cz, false, false);

      float p = pr[t];
#pragma unroll
      for (int v = 0; v < 8; v++) {
        float mine  = gate_fn(d[v], isZ);    // Z-lane: 1-sigmoid(Zp); H-lane: tanh(Hp)
        float other = __shfl_xor(mine, 8);   // pair Z-lane <-> H-lane
        acc[v] = fmaf(p * mine, other, acc[v]);
      }
    }

    // relu + transpose to LDS: H[sample_row][channel]
    if (isZ) {
      int ch = lane & 7;
      int rbase = hi ? 8 : 0;
#pragma unroll
      for (int v = 0; v < 8; v++) {
        float h = acc[v];
        Hsh[wv][rbase + v][ch] = h > 0.f ? h : 0.f;
      }
    }
  }
  __syncthreads();

  if (valid) {
    const float* B2 = cst + 80;
    v8f d2 = {0.f, 0.f, 0.f, 0.f, 0.f, 0.f, 0.f, 0.f};
#pragma unroll
    for (int kk = 0; kk < 2; kk++) {
      int k0 = kk * 4;
      v2f a2, b2;
      a2.x = Hsh[wv][ncol][k0 + (hi ? 2 : 0)];   // A: M=lane&15, K split by half-wave
      a2.y = Hsh[wv][ncol][k0 + (hi ? 3 : 1)];
      b2.x = B2[(k0 + (hi ? 2 : 0)) * 16 + ncol];
      b2.y = B2[(k0 + (hi ? 3 : 1)) * 16 + ncol];
      d2 = __builtin_amdgcn_wmma_f32_16x16x4_f32(
          false, a2, false, b2, (short)0, d2, false, false);
    }

    if (ncol < TOUT) {
      float bias = ob[ncol];
#pragma unroll
      for (int v = 0; v < 8; v++) {
        int  rrow = (hi ? 8 : 0) + v;
        long s    = base + rrow;
        int  node = (int)(s >> 5);
        int  bb   = (int)(s & 31);
        out[((long)bb * n + node) * TOUT + ncol] = d2[v] + bias;
      }
    }
  }
}

// ---------------- launcher ----------------
extern "C" void kernel_launch(void* const* d_in, const int* in_sizes, int n_in,
                              void* d_out, int out_size, void* d_ws, size_t ws_size,
                              hipStream_t stream) {
  const float*     x   = (const float*)d_in[0];
  const long long* ei  = (const long long*)d_in[1];   // int64 per reference
  const float*     ew  = (const float*)d_in[2];
  const float*     att = (const float*)d_in[3];
  const float*     Wz  = (const float*)d_in[4];
  const float*     bz  = (const float*)d_in[5];
  // d_in[6], d_in[7] (Wr, br) are dead code: H0==0 kills the R gate
  const float*     Wh  = (const float*)d_in[8];
  const float*     bh  = (const float*)d_in[9];
  const float*     Lz  = (const float*)d_in[10];
  const float*     Lzb = (const float*)d_in[11];
  const float*     Lh  = (const float*)d_in[14];
  const float*     Lhb = (const float*)d_in[15];
  const float*     ow  = (const float*)d_in[16];
  const float*     ob  = (const float*)d_in[17];
  float*           out = (float*)d_out;

  int E = in_sizes[2];
  int n = in_sizes[0] / (TB * TFIN * TT);

  // Workspace: [dinv (n)] [Xagg (n*768)] [cst (256)]
  float* ws   = (float*)d_ws;
  long   xoff = ((long)n + 255) & ~255L;
  float* dinv = ws;
  float* xagg = ws + xoff;
  float* cst  = xagg + (long)n * NBF;

  k_init_deg<<<(n + 255) / 256, 256, 0, stream>>>(dinv, n);
  k_edge_deg<<<(E + 255) / 256, 256, 0, stream>>>(ei, ew, dinv, E);
  k_dinv<<<(n + 255) / 256, 256, 0, stream>>>(dinv, n);
  k_precompute<<<1, 64, 0, stream>>>(att, Wz, bz, Wh, bh, Lz, Lzb, Lh, Lhb, ow, cst);

  long tot = (long)n * NBF;
  k_init_xagg<<<(int)((tot + 255) / 256), 256, 0, stream>>>(x, dinv, xagg, n);
  k_edge_scatter<<<(E + 7) / 8, 256, 0, stream>>>(ei, ew, dinv, x, xagg, E, n);

  long tiles = ((long)n * TB + 15) / 16;
  k_cell<<<(int)((tiles + 7) / 8), 256, 0, stream>>>(xagg, cst, ob, out, n);
}